// Encoder_57208964383496
// MI455X (gfx1250) — compile-verified
//
#include <hip/hip_runtime.h>

// ---------------------------------------------------------------------------
// SGConv(K=5, IN_F=1, OUT_F=64) + per-node BatchNorm, fused for MI455X.
//
// Phase 1 (bandwidth-bound): 4x SpMV over 32M edges.
//   - edge streams (src/dst/w, 1.5 GB total) read with NON-TEMPORAL loads so
//     they don't evict the 4 MB/hop node vectors from the 192 MB L2.
//   - scatter-add via hardware global_atomic_add_f32 (unsafeAtomicAdd).
// Phase 2 (write-bound): fused [N,5]x[5,64] GEMM + bias + BN + affine.
//   - V_WMMA_F32_16X16X4_F32: per wave, 16 nodes x 64 feats via 4 col-tiles
//     x 2 K-splits (K=5 padded to 8). Full f32, zero conversions.
//   - Fragment assembly is BRANCHLESS: lane-half selects are folded into
//     address arithmetic (k-offset = 2*(lane>>4)) or value cndmasks, so EXEC
//     never diverges and no s_and_saveexec appears near the WMMAs.
//   - BN stats (mean/var over the 64 feats of each node) reduced in-register
//     via shfl_xor within 16-lane halves; affine folded into one FMA.
//   - Output (256 MB, write-once) stored NON-TEMPORALLY: bigger than the
//     192 MB L2 and never re-read, so don't let it rinse the cache.
// ---------------------------------------------------------------------------

typedef __attribute__((ext_vector_type(2))) float v2f;
typedef __attribute__((ext_vector_type(8))) float v8f;

#define KHOPS  5
#define OUTF   64
#define BN_EPS 1e-5f

// ---- Phase 0: feats[0] = x, feats[1..4] = 0 (atomic accumulation targets) --
__global__ void init_feats_kernel(const float* __restrict__ x,
                                  float* __restrict__ f, int n) {
  int i = blockIdx.x * blockDim.x + threadIdx.x;
  int stride = gridDim.x * blockDim.x;
  for (; i < n; i += stride) {
    f[i] = x[i];
    f[(size_t)n + i]     = 0.f;
    f[(size_t)2 * n + i] = 0.f;
    f[(size_t)3 * n + i] = 0.f;
    f[(size_t)4 * n + i] = 0.f;
  }
}

// ---- Phase 1: one hop: next[dst] += prev[src] * w -------------------------
__global__ __launch_bounds__(256)
void spmv_kernel(const int* __restrict__ src, const int* __restrict__ dst,
                 const float* __restrict__ ew, const float* __restrict__ prev,
                 float* __restrict__ next, int e) {
  int i = blockIdx.x * blockDim.x + threadIdx.x;
  int stride = gridDim.x * blockDim.x;
  for (; i < e; i += stride) {
    // Stream the 384 MB/pass edge arrays non-temporally: keep L2 for the
    // 4 MB gather/scatter vectors.
    int   s = __builtin_nontemporal_load(&src[i]);
    int   d = __builtin_nontemporal_load(&dst[i]);
    float w = __builtin_nontemporal_load(&ew[i]);
    // Prefetch the edge stream one grid-stride ahead (global_prefetch_b8).
    __builtin_prefetch(&src[i + stride], 0, 0);
    float v = prev[s] * w;            // gather hits L2 (4 MB resident)
    unsafeAtomicAdd(&next[d], v);     // global_atomic_add_f32, no return
  }
}

// ---- Phase 2: fused GEMM(16x64 tile via WMMA) + bias + BN + affine --------
__global__ __launch_bounds__(256)
void gemm_bn_kernel(const float* __restrict__ feats,  // [5][N] column-major
                    const float* __restrict__ Wm,     // [64][5] (OUT_F,K)
                    const float* __restrict__ bias,   // [64]
                    const float* __restrict__ gamma,  // [N]
                    const float* __restrict__ beta,   // [N]
                    float* __restrict__ out, int n) { // [N][64]
  const int lane  = threadIdx.x & 31;
  const int wave  = threadIdx.x >> 5;
  const int tile  = blockIdx.x * (blockDim.x >> 5) + wave;
  const int ntile = n >> 4;                // N is a multiple of 16
  if (tile >= ntile) return;               // wave-uniform: EXEC stays all-1s
  const int base  = tile << 4;
  const int row16 = lane & 15;
  const int hi2   = lane >> 4;             // 0 for lanes 0-15, 1 for 16-31
  const int node  = base + row16;

  // A frags (16x4 f32): lanes 0-15 -> K=0,1 ; lanes 16-31 -> K=2,3.
  // Branchless: feature column index derived from lane half; loads are
  // unconditional with computed addresses, zero-pad done via value select.
  const float fa = feats[(size_t)(2 * hi2)     * n + node];  // F0 / F2
  const float fb = feats[(size_t)(2 * hi2 + 1) * n + node];  // F1 / F3
  const float f4 = feats[(size_t)4 * n + node];              // F4 (lo only)
  v2f a0, a1;
  a0.x = fa;
  a0.y = fb;
  a1.x = hi2 ? 0.f : f4;                   // v_cndmask, not a branch
  a1.y = 0.f;

  v8f acc[4];
#pragma unroll
  for (int t = 0; t < 4; ++t) {
    const int o = t * 16 + row16;          // output feature for this lane col
    // B frags (4x16 f32): VGPR0 holds rows K=0(lo)/K=2(hi), VGPR1 K=1/K=3.
    const float wb0 = Wm[o * KHOPS + 2 * hi2];
    const float wb1 = Wm[o * KHOPS + 2 * hi2 + 1];
    const float w4  = Wm[o * KHOPS + 4];
    v2f b0, b1;
    b0.x = wb0;
    b0.y = wb1;
    b1.x = hi2 ? 0.f : w4;                 // v_cndmask, not a branch
    b1.y = 0.f;

    v8f c = {};
    c = __builtin_amdgcn_wmma_f32_16x16x4_f32(false, a0, false, b0,
                                              (short)0, c, false, false);
    c = __builtin_amdgcn_wmma_f32_16x16x4_f32(false, a1, false, b1,
                                              (short)0, c, false, false);
    const float bo = bias[o];              // bias before BN stats (matters!)
#pragma unroll
    for (int r = 0; r < 8; ++r) c[r] += bo;
    acc[t] = c;
  }

  // BN over the 64 feats of each node. VGPR r holds row r (lanes 0-15) and
  // row r+8 (lanes 16-31); xor masks 1,2,4,8 reduce within each 16-lane half.
  const float inv64 = 1.0f / 64.0f;
#pragma unroll
  for (int r = 0; r < 8; ++r) {
    float p = acc[0][r] + acc[1][r] + acc[2][r] + acc[3][r];
    float q = acc[0][r] * acc[0][r] + acc[1][r] * acc[1][r] +
              acc[2][r] * acc[2][r] + acc[3][r] * acc[3][r];
#pragma unroll
    for (int m = 8; m >= 1; m >>= 1) {
      p += __shfl_xor(p, m, 32);
      q += __shfl_xor(q, m, 32);
    }
    const float mean = p * inv64;
    const float var  = q * inv64 - mean * mean;  // biased, matches reference
    const float rstd = rsqrtf(var + BN_EPS);
    const int   nd   = base + r + (hi2 << 3);
    const float g  = gamma[nd] * rstd;
    const float b2 = beta[nd] - mean * g;        // fold (v-mean)*rstd*g+beta
#pragma unroll
    for (int t = 0; t < 4; ++t) {
      // Write-once 256 MB stream: non-temporal, don't rinse L2.
      __builtin_nontemporal_store(acc[t][r] * g + b2,
                                  &out[(size_t)nd * OUTF + t * 16 + row16]);
    }
  }
}

extern "C" void kernel_launch(void* const* d_in, const int* in_sizes, int n_in,
                              void* d_out, int out_size, void* d_ws,
                              size_t ws_size, hipStream_t stream) {
  const float* x     = (const float*)d_in[0];  // [N,1]
  const int*   eidx  = (const int*)d_in[1];    // [2,E]
  const float* ew    = (const float*)d_in[2];  // [E]
  const float* Wm    = (const float*)d_in[3];  // [64,5,1]
  const float* bias  = (const float*)d_in[4];  // [64]
  const float* gamma = (const float*)d_in[5];  // [N]
  const float* beta  = (const float*)d_in[6];  // [N]
  float*       out   = (float*)d_out;          // [1,N,64]

  const int n = in_sizes[0];                   // N (IN_F == 1)
  const int e = in_sizes[2];                   // E
  const int* src = eidx;
  const int* dst = eidx + e;

  float* feats = (float*)d_ws;                 // [5][N] = 20 MB scratch

  init_feats_kernel<<<(n + 255) / 256, 256, 0, stream>>>(x, feats, n);

  for (int h = 1; h < KHOPS; ++h) {
    spmv_kernel<<<8192, 256, 0, stream>>>(src, dst, ew,
                                          feats + (size_t)(h - 1) * n,
                                          feats + (size_t)h * n, e);
  }

  const int ntile = n >> 4;                    // 16 nodes per wave
  const int waves_per_block = 8;               // 256 threads = 8 wave32
  gemm_bn_kernel<<<(ntile + waves_per_block - 1) / waves_per_block, 256, 0,
                   stream>>>(feats, Wm, bias, gamma, beta, out, n);
}